// InterpolateBilinear_20942260535437
// MI455X (gfx1250) — compile-verified
//
#include <hip/hip_runtime.h>

// ---------------------------------------------------------------------------
// tfa.image.interpolate_bilinear (indexing='ij') for
//   grid [B=8, H=256, W=256, C=64] f32, query_points [B, Q=65536, 2] (y,x) f32
//   -> out [B, Q, C] f32
//
// Memory-bound gather kernel (~0.02 flops/byte; HBM floor ~12us at 23.3TB/s).
// 16 lanes cooperate per query, each lane owns 4 channels (float4):
//   - 4x global_load_b128 corner gathers (TL/TR and BL/BR are contiguous
//     512B spans), RT temporal hint -> 134MB grid stays resident in 192MB L2
//   - all addressing as unsigned 32-bit v4f-element indices so codegen uses
//     the SADDR + voffset(scale_offset) form + immediate instruction offsets
//     (no 64-bit VALU address chains)
//   - non-temporal b128 store for the write-once 134MB output so it doesn't
//     evict the L2-resident grid
// ---------------------------------------------------------------------------

typedef float v4f __attribute__((ext_vector_type(4)));
typedef float v2f __attribute__((ext_vector_type(2)));

namespace {
constexpr int kB = 8;
constexpr int kH = 256;
constexpr int kW = 256;
constexpr int kC = 64;                  // floats per pixel
constexpr int kC4 = kC / 4;             // v4f per pixel = 16
constexpr int kQ = 65536;               // queries per batch image (2^16)
constexpr int kLanesPerQuery = 16;      // kC / 4
constexpr int kBlock = 256;             // 8 wave32 per block
}

__global__ __launch_bounds__(kBlock)
void InterpolateBilinear_20942260535437_kernel(const v4f* __restrict__ gridv,
                                               const v2f* __restrict__ qpv,
                                               v4f* __restrict__ outv) {
  const unsigned tid  = blockIdx.x * kBlock + threadIdx.x;
  const unsigned lane = tid & (kLanesPerQuery - 1);  // channel group (4 floats)
  const unsigned qidx = tid >> 4;                    // flat query id in [0, B*Q)
  const unsigned b    = qidx >> 16;                  // kQ == 2^16

  // (y, x); the 16 lanes of a query issue identical 8B loads (merged in the
  // memory pipe). 32-bit index + scale_offset addressing.
  const v2f yx = qpv[qidx];
  const float y = yx.x;
  const float x = yx.y;

  // Clamp floors so all four corners are in bounds (tfa semantics).
  const float y0f = fminf(fmaxf(floorf(y), 0.0f), (float)(kH - 2));
  const float x0f = fminf(fmaxf(floorf(x), 0.0f), (float)(kW - 2));
  const float ay  = fminf(fmaxf(y - y0f, 0.0f), 1.0f);
  const float ax  = fminf(fmaxf(x - x0f, 0.0f), 1.0f);
  const unsigned y0 = (unsigned)(int)y0f;
  const unsigned x0 = (unsigned)(int)x0f;

  // v4f-element index of this lane's TL corner chunk; max value
  // 8*256*256*16 = 2^23, so unsigned 32-bit -> SADDR + voffset codegen.
  const unsigned base = ((b * kH + y0) * kW + x0) * kC4 + lane;

  // Neighbor corners are constant element offsets: fold into the b128 load's
  // immediate offset. Four independent gathers issued back-to-back.
  const v4f tl = gridv[base];
  const v4f tr = gridv[base + kC4];
  const v4f bl = gridv[base + kW * kC4];
  const v4f br = gridv[base + kW * kC4 + kC4];

  // Exact reference arithmetic: top/bottom lerp in x, then lerp in y.
  const v4f top = tl + (tr - tl) * ax;
  const v4f bot = bl + (br - bl) * ax;
  const v4f res = top + (bot - top) * ay;

  // Streaming write-once output: non-temporal (TH=NT) store.
  __builtin_nontemporal_store(res, outv + qidx * (unsigned)kC4 + lane);
}

extern "C" void kernel_launch(void* const* d_in, const int* in_sizes, int n_in,
                              void* d_out, int out_size, void* d_ws, size_t ws_size,
                              hipStream_t stream) {
  (void)in_sizes; (void)n_in; (void)out_size; (void)d_ws; (void)ws_size;

  const v4f* gridv = (const v4f*)d_in[0];  // [B, H, W, C]   as v4f chunks
  const v2f* qpv   = (const v2f*)d_in[1];  // [B, Q, 2]      as (y,x) pairs
  v4f* outv        = (v4f*)d_out;          // [B, Q, C]      as v4f chunks

  constexpr long long total_threads =
      (long long)kB * kQ * kLanesPerQuery;               // 8,388,608
  constexpr int nblocks = (int)(total_threads / kBlock); // 32768, exact

  InterpolateBilinear_20942260535437_kernel<<<nblocks, kBlock, 0, stream>>>(
      gridv, qpv, outv);
}